// H2GCN_23390391894791
// MI455X (gfx1250) — compile-verified
//
#include <hip/hip_runtime.h>

typedef float v2f __attribute__((ext_vector_type(2)));
typedef float v8f __attribute__((ext_vector_type(8)));

#define KDIM   500
#define HIDDEN 64

// ---------------------------------------------------------------------------
// Stage 1: h = relu(x @ W1^T)   [N,500] x [64,500]^T -> [N,64]
// One wave per 16-row M-tile; the wave accumulates all 4 N-tiles (64 cols)
// so x is streamed from HBM exactly once (200 MB -> ~8.6us at 23.3 TB/s).
// x loads are non-temporal: single-use stream, keep L2 free for h + edges.
// ---------------------------------------------------------------------------
__global__ __launch_bounds__(256) void k_gemm_relu(
    const float* __restrict__ x, const float* __restrict__ W1,
    float* __restrict__ h, int n_tiles)
{
    const int lane   = threadIdx.x & 31;
    const int m_tile = blockIdx.x * 8 + (threadIdx.x >> 5);
    if (m_tile >= n_tiles) return;                 // wave-uniform: EXEC all-1s below
    const int nn     = lane & 15;                  // M index for A, N index for B/D
    const int khalf  = (lane >> 4) << 1;           // 0 or 2 (A/B K-layout)

    const float* xrow = x  + (size_t)(m_tile * 16 + nn) * KDIM;
    const float* w0   = W1 + (size_t)(nn +  0) * KDIM;
    const float* w1   = W1 + (size_t)(nn + 16) * KDIM;
    const float* w2   = W1 + (size_t)(nn + 32) * KDIM;
    const float* w3   = W1 + (size_t)(nn + 48) * KDIM;

    v8f acc0 = {}, acc1 = {}, acc2 = {}, acc3 = {};
    for (int k0 = 0; k0 < KDIM; k0 += 4) {
        const int k = k0 + khalf;
        v2f a  = __builtin_nontemporal_load((const v2f*)(xrow + k)); // stream x: NT
        v2f b0 = *(const v2f*)(w0 + k);                              // W1: L2-hot, RT
        v2f b1 = *(const v2f*)(w1 + k);
        v2f b2 = *(const v2f*)(w2 + k);
        v2f b3 = *(const v2f*)(w3 + k);
        acc0 = __builtin_amdgcn_wmma_f32_16x16x4_f32(false, a, false, b0, (short)0, acc0, false, false);
        acc1 = __builtin_amdgcn_wmma_f32_16x16x4_f32(false, a, false, b1, (short)0, acc1, false, false);
        acc2 = __builtin_amdgcn_wmma_f32_16x16x4_f32(false, a, false, b2, (short)0, acc2, false, false);
        acc3 = __builtin_amdgcn_wmma_f32_16x16x4_f32(false, a, false, b3, (short)0, acc3, false, false);
    }
    // D layout: M = r + 8*(lane>=16), N = lane&15
    const int m_lo = m_tile * 16 + ((lane >> 4) << 3);
    #pragma unroll
    for (int r = 0; r < 8; ++r) {
        const size_t mrow = (size_t)(m_lo + r) * HIDDEN;
        float v0 = acc0[r]; v0 = v0 > 0.f ? v0 : 0.f;
        float v1 = acc1[r]; v1 = v1 > 0.f ? v1 : 0.f;
        float v2 = acc2[r]; v2 = v2 > 0.f ? v2 : 0.f;
        float v3 = acc3[r]; v3 = v3 > 0.f ? v3 : 0.f;
        h[mrow + nn +  0] = v0;   // h is reused by SpMM + stage 3: keep RT (L2-resident)
        h[mrow + nn + 16] = v1;
        h[mrow + nn + 32] = v2;
        h[mrow + nn + 48] = v3;
    }
}

// ---------------------------------------------------------------------------
// Zero h1/h2 accumulators (float4 stores; re-read by atomics -> RT).
// ---------------------------------------------------------------------------
__global__ __launch_bounds__(256) void k_zero(float4* __restrict__ p, int n4)
{
    int i = blockIdx.x * blockDim.x + threadIdx.x;
    if (i < n4) p[i] = make_float4(0.f, 0.f, 0.f, 0.f);
}

// ---------------------------------------------------------------------------
// COO SpMM: out[dst] += val * h[src].  One wave per edge, 2 channels/lane.
// h rows hit L2 (25.6 MB << 192 MB); edge arrays streamed once -> NT loads;
// scatter via native relaxed agent-scope f32 atomics (global_atomic_add_f32).
// ---------------------------------------------------------------------------
__global__ __launch_bounds__(256) void k_spmm(
    const int* __restrict__ src, const int* __restrict__ dst,
    const float* __restrict__ val, const float* __restrict__ h,
    float* __restrict__ out, int nnz)
{
    const long long tid = (long long)blockIdx.x * blockDim.x + threadIdx.x;
    const int e = (int)(tid >> 5);
    if (e >= nnz) return;
    const int c = (int)((tid & 31) << 1);
    const int s = __builtin_nontemporal_load(src + e);
    const int d = __builtin_nontemporal_load(dst + e);
    const float v = __builtin_nontemporal_load(val + e);
    const v2f hv = *(const v2f*)(h + (size_t)s * HIDDEN + c);
    float* o = out + (size_t)d * HIDDEN + c;
    __hip_atomic_fetch_add(o,     v * hv.x, __ATOMIC_RELAXED, __HIP_MEMORY_SCOPE_AGENT);
    __hip_atomic_fetch_add(o + 1, v * hv.y, __ATOMIC_RELAXED, __HIP_MEMORY_SCOPE_AGENT);
}

// ---------------------------------------------------------------------------
// Stage 3: out = concat(h, h1, h2) @ Wf^T   (K=192, N=10 padded to 16)
// Lanes with n>=10 contribute B=0 (masked, branch-free -> EXEC all 1s).
// h/h1/h2 reads are last-use -> NT; output store is single-use -> NT.
// ---------------------------------------------------------------------------
__global__ __launch_bounds__(256) void k_final(
    const float* __restrict__ h,  const float* __restrict__ h1,
    const float* __restrict__ h2, const float* __restrict__ Wf,
    float* __restrict__ out, int n_tiles)
{
    const int lane = threadIdx.x & 31;
    const int tile = blockIdx.x * 8 + (threadIdx.x >> 5);
    if (tile >= n_tiles) return;                   // wave-uniform
    const int nn    = lane & 15;
    const int khalf = (lane >> 4) << 1;
    const int   wn    = nn < 10 ? nn : 9;
    const float wmask = nn < 10 ? 1.f : 0.f;
    const size_t rowoff = (size_t)(tile * 16 + nn) * HIDDEN;
    const float* wrow   = Wf + (size_t)wn * 192;

    const float* bufs[3] = { h, h1, h2 };
    v8f acc = {};
    #pragma unroll
    for (int seg = 0; seg < 3; ++seg) {
        const float* buf = bufs[seg] + rowoff;
        #pragma unroll
        for (int k0 = 0; k0 < 64; k0 += 4) {
            v2f a = __builtin_nontemporal_load((const v2f*)(buf + k0 + khalf));
            v2f b;
            b.x = wrow[seg * 64 + k0 + khalf + 0] * wmask;
            b.y = wrow[seg * 64 + k0 + khalf + 1] * wmask;
            acc = __builtin_amdgcn_wmma_f32_16x16x4_f32(false, a, false, b, (short)0, acc, false, false);
        }
    }
    if (nn < 10) {
        const int m_lo = tile * 16 + ((lane >> 4) << 3);
        #pragma unroll
        for (int r = 0; r < 8; ++r)
            __builtin_nontemporal_store(acc[r], out + (size_t)(m_lo + r) * 10 + nn);
    }
}

// ---------------------------------------------------------------------------
extern "C" void kernel_launch(void* const* d_in, const int* in_sizes, int n_in,
                              void* d_out, int out_size, void* d_ws, size_t ws_size,
                              hipStream_t stream)
{
    const float* x    = (const float*)d_in[0];
    const float* W1   = (const float*)d_in[1];
    const float* Wf   = (const float*)d_in[2];
    const int*   src1 = (const int*)  d_in[3];
    const int*   dst1 = (const int*)  d_in[4];
    const float* val1 = (const float*)d_in[5];
    const int*   src2 = (const int*)  d_in[6];
    const int*   dst2 = (const int*)  d_in[7];
    const float* val2 = (const float*)d_in[8];

    const int n    = in_sizes[0] / KDIM;   // 100000 nodes
    const int nnz1 = in_sizes[3];          // 600000
    const int nnz2 = in_sizes[6];          // 1200000

    float* h   = (float*)d_ws;
    float* h1  = h  + (size_t)n * HIDDEN;
    float* h2  = h1 + (size_t)n * HIDDEN;
    float* out = (float*)d_out;

    const int n_tiles = (n + 15) / 16;     // 6250 (exact)

    // Stage 1: dense GEMM + ReLU via fp32 WMMA (x streamed NT)
    k_gemm_relu<<<(n_tiles + 7) / 8, 256, 0, stream>>>(x, W1, h, n_tiles);

    // Zero h1 and h2 (contiguous region)
    {
        const long long n4 = (long long)2 * n * HIDDEN / 4;
        k_zero<<<(int)((n4 + 255) / 256), 256, 0, stream>>>((float4*)h1, (int)n4);
    }

    // Stage 2: two SpMMs via L2-resident gather + native f32 atomics
    {
        const long long t1 = (long long)nnz1 * 32;
        k_spmm<<<(int)((t1 + 255) / 256), 256, 0, stream>>>(src1, dst1, val1, h, h1, nnz1);
        const long long t2 = (long long)nnz2 * 32;
        k_spmm<<<(int)((t2 + 255) / 256), 256, 0, stream>>>(src2, dst2, val2, h, h2, nnz2);
    }

    // Stage 3: fused concat + final projection via fp32 WMMA
    k_final<<<(n_tiles + 7) / 8, 256, 0, stream>>>(h, h1, h2, Wf, out, n_tiles);
}